// BioSphericalCKN1D_54288386621702
// MI455X (gfx1250) — compile-verified
//
#include <hip/hip_runtime.h>
#include <hip/hip_bf16.h>

typedef __attribute__((ext_vector_type(2))) float v2f;
typedef __attribute__((ext_vector_type(8))) float v8f;
typedef __attribute__((ext_vector_type(4))) int   v4i;

#define B_   64
#define L_   16384
#define C_   20
#define F_   32
#define KW_  7
#define LO_  (L_ - KW_ + 1)          // 16378
#define KC_  (KW_ * C_)              // 140 = 35 * 4
#define KSTEPS 35
#define POS_PER_BLK 128              // 8 waves * 16 rows
#define XS_ROWS (POS_PER_BLK + KW_ - 1)   // 134
#define XS_FLOATS (XS_ROWS * C_)          // 2680 (divisible by 4)
#define KSW_PAIRS (KSTEPS * 2 * F_)       // 2240 float2 = 4480 floats
#define EPS_ 1e-7f

__global__ __launch_bounds__(256) void ckn1d_wmma_f32(
    const float* __restrict__ x,     // [B, L, C]
    const float* __restrict__ kw,    // [K, C, F]
    const float* __restrict__ sp,    // [1]
    const float* __restrict__ bp,    // [F]
    float* __restrict__ out)         // [B, LO, F]
{
    __shared__ float  xs[XS_FLOATS];
    __shared__ float2 ksw[KSW_PAIRS];   // B fragments pre-swizzled: adjacent {k[kr][n], k[kr+1][n]}

    const int tid   = threadIdx.x;
    const int batch = blockIdx.y;
    const int p0    = blockIdx.x * POS_PER_BLK;

    // ---- stage kernel weights, swizzled into WMMA-B fragment order ----
    // pair index t = (i*2 + h)*32 + n  ->  { kw[(4i+2h)*F + n], kw[(4i+2h+1)*F + n] }
    #pragma unroll 2
    for (int t = tid; t < KSW_PAIRS; t += 256) {
        const int i   = t >> 6;          // kstep
        const int rem = t & 63;
        const int h   = rem >> 5;        // lane half
        const int n   = rem & 31;        // column
        const int r   = 4 * i + 2 * h;   // flat K row (t*C + c)
        float2 v;
        v.x = kw[r * F_ + n];
        v.y = kw[(r + 1) * F_ + n];
        ksw[t] = v;
    }

    // ---- stage x slab (contiguous, clamped at image end) ----
    {
        const int per_batch4 = (L_ * C_) / 4;          // 81920
        const float4* x4  = (const float4*)(x + (size_t)batch * L_ * C_);
        float4*       xs4 = (float4*)xs;
        const int base4 = p0 * (C_ / 4);               // p0 * 5
#if defined(__gfx1250__) && __has_builtin(__builtin_amdgcn_global_load_async_to_lds_b128)
        // CDNA5 async direct-to-LDS path (ASYNCcnt-tracked, bypasses VGPRs)
        #pragma unroll 2
        for (int t = tid; t < XS_FLOATS / 4; t += 256) {
            int i4 = base4 + t;
            if (i4 > per_batch4 - 1) i4 = per_batch4 - 1;
            __builtin_amdgcn_global_load_async_to_lds_b128(
                (__attribute__((address_space(1))) v4i*)(x4 + i4),
                (__attribute__((address_space(3))) v4i*)(xs4 + t),
                0, 0);
        }
#if __has_builtin(__builtin_amdgcn_s_wait_asynccnt)
        __builtin_amdgcn_s_wait_asynccnt(0);
#else
        asm volatile("s_wait_asynccnt 0" ::: "memory");
#endif
#else
        #pragma unroll 2
        for (int t = tid; t < XS_FLOATS / 4; t += 256) {
            int i4 = base4 + t;
            if (i4 > per_batch4 - 1) i4 = per_batch4 - 1;
            xs4[t] = x4[i4];
        }
#endif
    }
    __syncthreads();

    const int wave   = tid >> 5;     // 0..7
    const int lane   = tid & 31;
    const int laneLo = lane & 15;
    const int hi     = lane >> 4;    // 0: lanes 0-15, 1: lanes 16-31

    const int m0  = wave * 16;       // LDS row base of this wave's M-tile
    // A-matrix row for this lane: ds_load_b64 at (row*20 + 4i + 2*hi), 8B-aligned
    const float2* arow = (const float2*)(xs + (m0 + laneLo) * C_ + 2 * hi);
    // B fragments for this lane: adjacent float2 per kstep; b1 = b0 + 16 pairs (128 B)
    const float2* brow = ksw + hi * F_ + laneLo;

    v8f acc0 = {};                   // N = 0..15
    v8f acc1 = {};                   // N = 16..31
    float ssq_part = 0.0f;

    #pragma unroll
    for (int i = 0; i < KSTEPS; ++i) {
        float2 a2 = arow[2 * i];                 // A[row][4i+2hi .. +1]
        ssq_part = __builtin_fmaf(a2.x, a2.x, ssq_part);
        ssq_part = __builtin_fmaf(a2.y, a2.y, ssq_part);

        float2 f0 = brow[i * 2 * F_];            // cols 0..15  -> ds_load_2addr_b64
        float2 f1 = brow[i * 2 * F_ + 16];       // cols 16..31 /

        v2f av; av.x = a2.x; av.y = a2.y;
        v2f b0; b0.x = f0.x; b0.y = f0.y;
        v2f b1; b1.x = f1.x; b1.y = f1.y;

        acc0 = __builtin_amdgcn_wmma_f32_16x16x4_f32(false, av, false, b0,
                                                     (short)0, acc0, false, false);
        acc1 = __builtin_amdgcn_wmma_f32_16x16x4_f32(false, av, false, b1,
                                                     (short)0, acc1, false, false);
    }

    // ---- fused window L2-norm: lane L holds K%4 in {0,1}, lane L+16 holds {2,3}
    float ssq = ssq_part + __shfl_xor(ssq_part, 16, 32);
    const float sv = sp[0];
    const float rnorm = sv * __frsqrt_rn(ssq + EPS_);   // valid on all lanes, row = laneLo

    const float bias0 = bp[laneLo];
    const float bias1 = bp[16 + laneLo];

    // ---- scale + bias + store; D layout: vgpr j, lanes<16 -> M=j, lanes>=16 -> M=j+8
    float* obase = out + ((size_t)batch * LO_ + (size_t)(p0 + m0)) * F_;
    #pragma unroll
    for (int j = 0; j < 8; ++j) {
        const int row = 8 * hi + j;
        const int pos = p0 + m0 + row;
        if (pos < LO_) {
            const float rn = __shfl(rnorm, 8 * hi + j, 32);  // broadcast row's norm
            float* o = obase + (size_t)row * F_;
            o[laneLo]      = __builtin_fmaf(acc0[j], rn, bias0);
            o[16 + laneLo] = __builtin_fmaf(acc1[j], rn, bias1);
        }
    }
}

extern "C" void kernel_launch(void* const* d_in, const int* in_sizes, int n_in,
                              void* d_out, int out_size, void* d_ws, size_t ws_size,
                              hipStream_t stream) {
    (void)in_sizes; (void)n_in; (void)d_ws; (void)ws_size; (void)out_size;
    const float* x  = (const float*)d_in[0];
    const float* kw = (const float*)d_in[1];
    const float* sp = (const float*)d_in[2];
    const float* bp = (const float*)d_in[3];
    float* out = (float*)d_out;

    dim3 grid((LO_ + POS_PER_BLK - 1) / POS_PER_BLK, B_);   // 128 x 64
    dim3 block(256);
    ckn1d_wmma_f32<<<grid, block, 0, stream>>>(x, kw, sp, bp, out);
}